// Hbv_2_hourly_53506702574203
// MI455X (gfx1250) — compile-verified
//
#include <hip/hip_runtime.h>
#include <hip/hip_bf16.h>
#include <math.h>

// ---------------------------------------------------------------------------
// HBV hydrological model (sequential scan) + per-grid gamma routing conv.
//  * No dense matmul exists in this workload (depthwise conv, per-grid
//    filters) -> WMMA not applicable. CDNA5 paths used instead:
//      - TENSOR_LOAD_TO_LDS (TDM DMA, TENSORcnt) for steady-state conv tiles
//      - global_load_async_to_lds_b32 / s_wait_asynccnt (ASYNCcnt engine)
//      - global_prefetch (via __builtin_prefetch) in the scan hot loop
//      - LDS ring as indexed sliding-window storage (ds ops), wave32 blocks
// ---------------------------------------------------------------------------

#define NSTEPS 8760
#define NGRID  2048
#define NMUL   2
#define NPHY   19
#define LENF   72
#define DTF    (1.0f / 24.0f)

// ------------------------------ fast pow -----------------------------------
__device__ __forceinline__ float powx(float a, float b) {
  // a > 0 guaranteed by callers; v_log_f32 + v_exp_f32 (TRANS pipe)
  return __expf(b * __logf(a));
}

// ======================= Kernel 1: HBV scan ================================
// One thread per grid cell; both NMUL parameter sets kept in registers as two
// independent dependency chains (ILP). Forcing reads are coalesced (time-major
// layout). Prefetch 16 steps ahead -> global_prefetch_b8.
__global__ __launch_bounds__(256) void hbv_scan_kernel(
    const float* __restrict__ prcp, const float* __restrict__ tmean,
    const float* __restrict__ pet, const float* __restrict__ phy,
    float* __restrict__ q) {
  const int g = blockIdx.x * blockDim.x + threadIdx.x;
  if (g >= NGRID) return;

  static const float LB[NPHY] = {1.f, 50.f,  0.05f, 0.01f, 0.001f, 0.2f, 0.f,
                                 0.f, -2.5f, 0.5f,  0.f,   0.f,    0.3f, 0.f,
                                 0.f, 0.f,   120.f, 0.f,   0.5f};
  static const float UB[NPHY] = {6.f,  1000.f, 0.9f, 0.5f, 0.2f,    1.f,  10.f,
                                 100.f, 2.5f,  10.f, 0.1f, 0.2f,    5.f,  1.f,
                                 20.f,  2500.f, 2880.f, 1.f, 5.f};

  float BETA[2], FCv[2], invFC[2], K0dt[2], K1dt[2], K2dt[2], invLPFC[2],
      PERCdt[2], UZLv[2], TTv[2], CFMAXdt[2], CFRdt[2], CWHv[2], BETAET[2],
      Cc[2], F0dt[2], FMINv[2], OMF[2], ALPHAv[2];
#pragma unroll
  for (int m = 0; m < 2; ++m) {
    float pv[NPHY];
#pragma unroll
    for (int i = 0; i < NPHY; ++i)
      pv[i] = LB[i] + phy[(size_t)g * (NPHY * NMUL) + i * NMUL + m] * (UB[i] - LB[i]);
    BETA[m] = pv[0];
    FCv[m] = pv[1];
    invFC[m] = 1.f / pv[1];
    K0dt[m] = pv[2] * DTF;
    K1dt[m] = pv[3] * DTF;
    K2dt[m] = pv[4] * DTF;
    invLPFC[m] = 1.f / (pv[5] * pv[1]);
    PERCdt[m] = pv[6] * DTF;
    UZLv[m] = pv[7];
    TTv[m] = pv[8];
    CFMAXdt[m] = pv[9] * DTF;
    CFRdt[m] = pv[10] * pv[9] * DTF;
    CWHv[m] = pv[11];
    BETAET[m] = pv[12];
    Cc[m] = pv[13];
    F0dt[m] = pv[16] * DTF;
    FMINv[m] = pv[17];
    OMF[m] = 1.f - pv[17];
    ALPHAv[m] = pv[18];
  }

  float SP_[2], MW_[2], SM_[2], SU_[2], SL_[2];
#pragma unroll
  for (int m = 0; m < 2; ++m) SP_[m] = MW_[m] = SM_[m] = SU_[m] = SL_[m] = 0.001f;

  for (int t = 0; t < NSTEPS; ++t) {
    const size_t off = (size_t)t * NGRID + g;
    const float P = prcp[off];
    const float T = tmean[off];
    const float PT = pet[off];
    if (t + 16 < NSTEPS) {  // -> global_prefetch_b8 (speculative RT)
      const size_t o2 = off + (size_t)16 * NGRID;
      __builtin_prefetch(&prcp[o2], 0, 3);
      __builtin_prefetch(&tmean[o2], 0, 3);
      __builtin_prefetch(&pet[o2], 0, 3);
    }
    float Qs = 0.f;
#pragma unroll
    for (int m = 0; m < 2; ++m) {
      float SP = SP_[m], MW = MW_[m], SM = SM_[m], SU = SU_[m], SL = SL_[m];
      const float rain = (T >= TTv[m]) ? P : 0.f;
      const float snow = (T < TTv[m]) ? P : 0.f;
      SP += snow;
      const float melt = fminf(fmaxf(CFMAXdt[m] * (T - TTv[m]), 0.f), SP);
      MW += melt;
      SP -= melt;
      const float refr = fminf(fmaxf(CFRdt[m] * (TTv[m] - T), 0.f), MW);
      SP += refr;
      MW -= refr;
      const float tosoil = fmaxf(MW - CWHv[m] * SP, 0.f);
      MW -= tosoil;
      const float win = rain + tosoil;
      const float sat = fminf(fmaxf(SM * invFC[m], 0.f), 1.f);
      const float fcap =
          F0dt[m] * (FMINv[m] + OMF[m] * powx(fmaxf(1.f - sat, 1e-6f), ALPHAv[m]));
      const float infil = fminf(win, fcap);
      const float sexcs = win - infil;
      const float swet = fminf(fmaxf(powx(SM * invFC[m], BETA[m]), 0.f), 1.f);
      const float rech = infil * swet;
      SM += infil - rech;
      const float excs = fmaxf(SM - FCv[m], 0.f);
      SM -= excs;
      const float ef = fminf(fmaxf(powx(SM * invLPFC[m], BETAET[m]), 0.f), 1.f);
      const float aet = fminf(SM, PT * ef);
      SM = fmaxf(SM - aet, 1e-5f);
      const float cap = fminf(SL, Cc[m] * SL * (1.f - fminf(SM * invFC[m], 1.f)));
      SM += cap;
      SL -= cap;
      SU += rech + excs + sexcs;
      const float perc = fminf(SU, PERCdt[m]);
      SU -= perc;
      SL += perc;
      const float Q0 = K0dt[m] * fmaxf(SU - UZLv[m], 0.f);
      SU -= Q0;
      const float Q1 = K1dt[m] * SU;
      SU -= Q1;
      const float Q2 = K2dt[m] * SL;
      SL -= Q2;
      Qs += Q0 + Q1 + Q2;
      SP_[m] = SP; MW_[m] = MW; SM_[m] = SM; SU_[m] = SU; SL_[m] = SL;
    }
    q[off] = 0.5f * Qs;  // mean over NMUL=2
  }
}

// ===================== Kernel 2: gamma routing conv ========================
#define GPB      64   // grids per block (block = 64 threads = 2 waves)
#define RINGROWS 192  // LDS ring depth = 4*CHUNK; window + prefetch <= 167
#define CHUNK    48
#define TTILE    480  // time rows per block tile (split path)

__device__ __forceinline__ unsigned slot_of(int r) {
  return (unsigned)(r + RINGROWS * 256) % (unsigned)RINGROWS;
}

// CDNA5 async copy engine: global -> LDS, tracked with ASYNCcnt.
__device__ __forceinline__ void async_g2l(const float* gp, float* lp) {
  const unsigned lds = (unsigned)(uintptr_t)lp;  // low 32 bits = LDS address
  asm volatile("global_load_async_to_lds_b32 %0, %1, off" ::"v"(lds), "v"(gp)
               : "memory");
}
__device__ __forceinline__ void wait_async0() {
  asm volatile("s_wait_asynccnt 0" ::: "memory");
}
__device__ __forceinline__ void wait_tensor0() {
#if __has_builtin(__builtin_amdgcn_s_wait_tensorcnt)
  __builtin_amdgcn_s_wait_tensorcnt(0);
#else
  asm volatile("s_wait_tensorcnt 0x0" ::: "memory");
#endif
}

// --------------------- Tensor Data Mover tile load -------------------------
// 2D tile: tile_dim0 = 64 floats (one LDS ring row), tile_dim1 = CHUNK rows,
// global row stride = NGRID. tensor_dim1 = #valid rows from the tile start so
// TDM's OOB rule zero-fills rows past NSTEPS. Wave-uniform SGPR descriptor
// groups (D# group0: 4 SGPRs, group1: 8 SGPRs); groups 2/3 = NULL (2D).
typedef unsigned int u32x4 __attribute__((ext_vector_type(4)));
typedef unsigned int u32x8 __attribute__((ext_vector_type(8)));

__device__ __forceinline__ void tdm_load_tile(const float* gsrc,
                                              unsigned lds_byte,
                                              unsigned valid_rows) {
  const unsigned long long ga = (unsigned long long)(uintptr_t)gsrc;
  u32x4 g0;
  g0[0] = 1u;                                   // count=1 (valid, user mode)
  g0[1] = lds_byte;                             // lds_addr (bytes)
  g0[2] = (unsigned)ga;                         // global_addr[31:0]
  g0[3] = (unsigned)(ga >> 32) | 0x80000000u;   // addr[56:32] | type=2
  u32x8 g1;
  g1[0] = 0x00020000u;                          // data_size=4B; no flags
  g1[1] = (unsigned)GPB << 16;                  // tensor_dim0 = 64
  g1[2] = (valid_rows & 0xffffu) << 16;         // tensor_dim1[15:0]
  g1[3] = (unsigned)GPB << 16;                  // tensor_dim1 hi=0 | tile_dim0=64
  g1[4] = (unsigned)CHUNK;                      // tile_dim1=48, tile_dim2=0
  g1[5] = (unsigned)NGRID;                      // tensor_dim0_stride = 2048
  g1[6] = 0u;                                   // stride0 hi | stride1 lo
  g1[7] = 0u;                                   // tensor_dim1_stride hi
  asm volatile("tensor_load_to_lds %0, %1" ::"s"(g0), "s"(g1) : "memory");
}

// Per-thread gamma filter (72 taps) straight into registers.
__device__ __forceinline__ void gamma_weights(const float* __restrict__ distr,
                                              int g, float (&w)[LENF]) {
  const float aa = fmaxf(distr[g * 3 + 0] * 5.0f, 0.f) + 0.1f;
  const float th = fmaxf(distr[g * 3 + 1] * 12.0f, 0.f) + 0.5f;
  const float tau = distr[g * 3 + 2] * 48.0f;
  const float c0 = -lgammaf(aa) - aa * __logf(th);
  const float am1 = aa - 1.f;
  const float ith = 1.f / th;
  float s = 0.f;
#pragma unroll
  for (int k = 0; k < LENF; ++k) {
    const float ts = fmaxf((float)k + 0.5f - tau, 0.001f);
    const float wv = __expf(c0 + am1 * __logf(ts) - ts * ith);
    w[k] = wv;
    s += wv;
  }
  const float inv = 1.f / s;
#pragma unroll
  for (int k = 0; k < LENF; ++k) w[k] *= inv;
}

// Convolve one chunk out of the LDS ring. Each thread reads only its own
// column (LDS is indexed storage for the 72-deep sliding window).
__device__ __forceinline__ void conv_chunk(const float (*ring)[GPB],
                                           const float (&w)[LENF], int tid,
                                           int t0, int tmax,
                                           float* __restrict__ out, int g) {
  for (int tt = t0; tt < tmax; tt += 4) {
    float a0 = 0.f, a1 = 0.f, a2 = 0.f, a3 = 0.f;
    unsigned s = slot_of(tt - (LENF - 1));
#pragma unroll
    for (int rr = 0; rr < LENF + 3; ++rr) {  // rows tt-71 .. tt+3
      const float v = ring[s][tid];
      s = (s + 1u == RINGROWS) ? 0u : s + 1u;
      int k;
      k = 71 - rr; if (k >= 0)            a0 += v * w[k];
      k = 72 - rr; if (k >= 0 && k < 72)  a1 += v * w[k];
      k = 73 - rr; if (k >= 0 && k < 72)  a2 += v * w[k];
      k = 74 - rr; if (k >= 0 && k < 72)  a3 += v * w[k];
    }
    if (tt + 0 < tmax) out[(size_t)(tt + 0) * NGRID + g] = a0;
    if (tt + 1 < tmax) out[(size_t)(tt + 1) * NGRID + g] = a1;
    if (tt + 2 < tmax) out[(size_t)(tt + 2) * NGRID + g] = a2;
    if (tt + 3 < tmax) out[(size_t)(tt + 3) * NGRID + g] = a3;
  }
}

// ---- split path: q in scratch, out separate; time tiled across blocks ----
__global__ __launch_bounds__(GPB) void conv_split_kernel(
    const float* __restrict__ q, const float* __restrict__ distr,
    float* __restrict__ out) {
  __shared__ float ring[RINGROWS][GPB];  // 48 KB
  const int tid = threadIdx.x;
  const int g = blockIdx.x * GPB + tid;
  const int tbeg = blockIdx.y * TTILE;
  const int tend = min(tbeg + TTILE, NSTEPS);

  float w[LENF];
  gamma_weights(distr, g, w);

  // Prologue: halo + first chunk (wraps the ring -> per-lane async engine).
  for (int r = tbeg - (LENF - 1); r <= tbeg + CHUNK - 1; ++r) {
    const unsigned s = slot_of(r);
    if (r >= 0 && r < NSTEPS)
      async_g2l(&q[(size_t)r * NGRID + g], &ring[s][tid]);
    else
      ring[s][tid] = 0.f;
  }
  wait_async0();
  __syncthreads();

  for (int t0 = tbeg; t0 < tend; t0 += CHUNK) {
    // Steady state: one TDM DMA stages the next 48x64 tile while we compute.
    // t0 multiples of 48 and RINGROWS = 4*48 -> chunk slots are contiguous,
    // and prefetch slots never alias the live read window mod 192.
    if (t0 + CHUNK < tend && tid < 32) {  // wave 0 issues; TDM is per-wave
      const int r0 = t0 + CHUNK;
      tdm_load_tile(&q[(size_t)r0 * NGRID + (size_t)blockIdx.x * GPB],
                    (unsigned)(uintptr_t)&ring[slot_of(r0)][0],
                    (unsigned)(NSTEPS - r0));
    }
    conv_chunk(ring, w, tid, t0, min(t0 + CHUNK, tend), out, g);
    wait_tensor0();   // wave0: DMA done; wave1: TENSORcnt==0, passes
    __syncthreads();  // publish wave0's tile to wave1
  }
}

// ---- fallback path: convolve d_out in place, descending time chunks ----
__global__ __launch_bounds__(GPB) void conv_inplace_kernel(
    float* __restrict__ qout, const float* __restrict__ distr) {
  __shared__ float ring[RINGROWS][GPB];
  const int tid = threadIdx.x;
  const int g = blockIdx.x * GPB + tid;

  float w[LENF];
  gamma_weights(distr, g, w);

  const int cLast = (NSTEPS + CHUNK - 1) / CHUNK - 1;  // 182
  int t0 = cLast * CHUNK;
  for (int r = t0 - (LENF - 1); r <= t0 + CHUNK - 1; ++r) {
    const unsigned s = slot_of(r);
    if (r >= 0 && r < NSTEPS)
      async_g2l(&qout[(size_t)r * NGRID + g], &ring[s][tid]);
    else
      ring[s][tid] = 0.f;
  }
  wait_async0();

  for (int c = cLast; c >= 0; --c) {
    t0 = c * CHUNK;
    conv_chunk(ring, w, tid, t0, min(t0 + CHUNK, NSTEPS), qout, g);
    if (c > 0) {
      // New halo rows for the next (lower) chunk: all strictly below any row
      // overwritten so far -> in-place safe; columns are thread-disjoint.
      for (int r = t0 - CHUNK - (LENF - 1); r <= t0 - LENF; ++r) {
        const unsigned s = slot_of(r);
        if (r >= 0)
          async_g2l(&qout[(size_t)r * NGRID + g], &ring[s][tid]);
        else
          ring[s][tid] = 0.f;
      }
      wait_async0();
    }
  }
}

// ============================== launcher ===================================
extern "C" void kernel_launch(void* const* d_in, const int* in_sizes, int n_in,
                              void* d_out, int out_size, void* d_ws,
                              size_t ws_size, hipStream_t stream) {
  (void)in_sizes; (void)n_in; (void)out_size;
  const float* prcp = (const float*)d_in[0];
  const float* tmean = (const float*)d_in[1];
  const float* pet = (const float*)d_in[2];
  const float* phy = (const float*)d_in[3];
  const float* distr = (const float*)d_in[4];
  float* out = (float*)d_out;

  const size_t qbytes = (size_t)NSTEPS * NGRID * sizeof(float);
  if (ws_size >= qbytes) {
    float* q = (float*)d_ws;
    hbv_scan_kernel<<<NGRID / 256, 256, 0, stream>>>(prcp, tmean, pet, phy, q);
    dim3 grid(NGRID / GPB, (NSTEPS + TTILE - 1) / TTILE);
    conv_split_kernel<<<grid, GPB, 0, stream>>>(q, distr, out);
  } else {
    hbv_scan_kernel<<<NGRID / 256, 256, 0, stream>>>(prcp, tmean, pet, phy, out);
    conv_inplace_kernel<<<NGRID / GPB, GPB, 0, stream>>>(out, distr);
  }
}